// HungarianMatcher_54546084659234
// MI455X (gfx1250) — compile-verified
//
#include <hip/hip_runtime.h>
#include <float.h>
#include <math.h>

typedef __attribute__((ext_vector_type(2))) float v2f;
typedef __attribute__((ext_vector_type(8))) float v8f;

#define NCLS      26
#define NCLS_PAD  28           // pad K to multiple of 4; padded probs are 0
#define MT        800          // number of targets
#define QT        16           // query rows per block (WMMA tile M)
#define NTILES    (MT / 16)    // 50 column tiles of 16
#define COST_CLASS 2.0f
#define COST_BBOX  5.0f
#define COST_GIOU  2.0f

// Fast reciprocal: single v_rcp_f32 (~1 ulp) instead of the full IEEE
// div_scale/rcp/fma-fixup expansion (~8-10 VALU ops per division).
__device__ __forceinline__ float fast_rcp(float x) {
    return __builtin_amdgcn_rcpf(x);
}

// min 4 waves per SIMD -> cap at 256 VGPRs/lane: avoids VGPR-MSB addressing
// (s_set_vgpr_msb churn) and keeps 2 blocks resident per WGP for latency hiding.
__global__ __launch_bounds__(256, 4) void matcher_cost_kernel(
    const float* __restrict__ pred_logits,   // [N, 26]
    const float* __restrict__ pred_boxes,    // [N, 6]  cxcyczwhd
    const float* __restrict__ tgt_boxes,     // [M, 6]  cxcyczwhd
    const int*   __restrict__ tgt_labels,    // [M]
    float* __restrict__ out)                 // [N, M]
{
    // Target SoA in LDS: raw(6) | lo(3) | hi(3) | vol(1) = 13 float arrays
    __shared__ float sm_t[13 * MT];          // 41600 B
    __shared__ int   sm_lab[MT];             //  3200 B
    __shared__ float sm_prob[QT * NCLS_PAD]; //  1792 B (softmax rows, zero-padded)
    __shared__ float sm_q[QT * 13];          //   832 B (raw6, lo3, hi3, vol)

    const int tid     = threadIdx.x;
    const int rowbase = blockIdx.x * QT;

    // ---------------- Phase 1a: stage + convert target table ----------------
    for (int m = tid; m < MT; m += 256) {
        float c0 = tgt_boxes[m * 6 + 0], c1 = tgt_boxes[m * 6 + 1], c2 = tgt_boxes[m * 6 + 2];
        float s0 = tgt_boxes[m * 6 + 3], s1 = tgt_boxes[m * 6 + 4], s2 = tgt_boxes[m * 6 + 5];
        sm_t[0 * MT + m] = c0;  sm_t[1 * MT + m] = c1;  sm_t[2 * MT + m] = c2;
        sm_t[3 * MT + m] = s0;  sm_t[4 * MT + m] = s1;  sm_t[5 * MT + m] = s2;
        float lo0 = c0 - 0.5f * s0, lo1 = c1 - 0.5f * s1, lo2 = c2 - 0.5f * s2;
        float hi0 = c0 + 0.5f * s0, hi1 = c1 + 0.5f * s1, hi2 = c2 + 0.5f * s2;
        sm_t[6 * MT + m] = lo0;  sm_t[7 * MT + m]  = lo1;  sm_t[8 * MT + m]  = lo2;
        sm_t[9 * MT + m] = hi0;  sm_t[10 * MT + m] = hi1;  sm_t[11 * MT + m] = hi2;
        sm_t[12 * MT + m] = (hi0 - lo0) * (hi1 - lo1) * (hi2 - lo2);
        sm_lab[m] = tgt_labels[m];
    }

    // ---------------- Phase 1b: per-row softmax (threads 0..15) -------------
    if (tid < QT) {
        const float* lg = pred_logits + (size_t)(rowbase + tid) * NCLS;
        float mx = -FLT_MAX;
        #pragma unroll
        for (int c = 0; c < NCLS; ++c) mx = fmaxf(mx, lg[c]);
        float s = 0.0f;
        #pragma unroll
        for (int c = 0; c < NCLS; ++c) {
            float e = expf(lg[c] - mx);
            sm_prob[tid * NCLS_PAD + c] = e;
            s += e;
        }
        float inv = fast_rcp(s);
        #pragma unroll
        for (int c = 0; c < NCLS; ++c) sm_prob[tid * NCLS_PAD + c] *= inv;
        sm_prob[tid * NCLS_PAD + NCLS + 0] = 0.0f;   // K padding -> contributes 0
        sm_prob[tid * NCLS_PAD + NCLS + 1] = 0.0f;
    }
    // ---------------- Phase 1c: query boxes (threads 16..31) ----------------
    else if (tid < 2 * QT) {
        int r = tid - QT;
        const float* pb = pred_boxes + (size_t)(rowbase + r) * 6;
        float c0 = pb[0], c1 = pb[1], c2 = pb[2], s0 = pb[3], s1 = pb[4], s2 = pb[5];
        float* q = &sm_q[r * 13];
        q[0] = c0; q[1] = c1; q[2] = c2; q[3] = s0; q[4] = s1; q[5] = s2;
        float lo0 = c0 - 0.5f * s0, lo1 = c1 - 0.5f * s1, lo2 = c2 - 0.5f * s2;
        float hi0 = c0 + 0.5f * s0, hi1 = c1 + 0.5f * s1, hi2 = c2 + 0.5f * s2;
        q[6] = lo0; q[7] = lo1; q[8] = lo2;
        q[9] = hi0; q[10] = hi1; q[11] = hi2;
        q[12] = (hi0 - lo0) * (hi1 - lo1) * (hi2 - lo2);
    }
    __syncthreads();

    // ---------------- Phase 2: 16x16 tiles, WMMA class cost + fused GIoU ----
    const int lane = tid & 31;
    const int wave = tid >> 5;
    const int hh   = lane >> 4;     // lane-half: 0 or 1
    const int l15  = lane & 15;

    for (int tile = wave; tile < NTILES; tile += 8) {
        const int colbase = tile * 16;
        const int m       = colbase + l15;     // this lane's target column
        const int lbl     = sm_lab[m];

        // --- class cost via exact one-hot f32 WMMA: acc = P_tile x OneHot ---
        v8f acc = {};
        #pragma unroll
        for (int kb = 0; kb < NCLS_PAD; kb += 4) {
            // A 16x4 f32: VGPR j, lane L -> row = L&15, K = kb + j + 2*(L>=16)
            int ka = kb + 2 * hh;
            v2f a, b;
            a.x = sm_prob[l15 * NCLS_PAD + ka + 0];
            a.y = sm_prob[l15 * NCLS_PAD + ka + 1];
            // B 4x16 one-hot: col = L&15, same K mapping
            b.x = (lbl == ka + 0) ? 1.0f : 0.0f;
            b.y = (lbl == ka + 1) ? 1.0f : 0.0f;
            acc = __builtin_amdgcn_wmma_f32_16x16x4_f32(
                false, a, false, b, (short)0, acc, false, false);
        }
        // acc[i] = prob[row = i + 8*hh][ lbl ]  at column m (exact)

        // --- target data for this lane's column, held in registers ---------
        float tc0 = sm_t[0 * MT + m], tc1 = sm_t[1 * MT + m], tc2 = sm_t[2 * MT + m];
        float ts0 = sm_t[3 * MT + m], ts1 = sm_t[4 * MT + m], ts2 = sm_t[5 * MT + m];
        float tl0 = sm_t[6 * MT + m], tl1 = sm_t[7 * MT + m],  tl2 = sm_t[8 * MT + m];
        float th0 = sm_t[9 * MT + m], th1 = sm_t[10 * MT + m], th2 = sm_t[11 * MT + m];
        float tvol = sm_t[12 * MT + m];

        #pragma unroll
        for (int i = 0; i < 8; ++i) {
            const int r = i + 8 * hh;              // local query row (C/D layout)
            const float* q = &sm_q[r * 13];
            // L1 over raw cxcyczwhd
            float l1 = fabsf(q[0] - tc0) + fabsf(q[1] - tc1) + fabsf(q[2] - tc2)
                     + fabsf(q[3] - ts0) + fabsf(q[4] - ts1) + fabsf(q[5] - ts2);
            // intersection
            float ix = fmaxf(fminf(q[9],  th0) - fmaxf(q[6], tl0), 0.0f);
            float iy = fmaxf(fminf(q[10], th1) - fmaxf(q[7], tl1), 0.0f);
            float iz = fmaxf(fminf(q[11], th2) - fmaxf(q[8], tl2), 0.0f);
            float inter = ix * iy * iz;
            float uni   = q[12] + tvol - inter;
            float iou   = inter * fast_rcp(uni);
            // enclosing box
            float ex = fmaxf(fmaxf(q[9],  th0) - fminf(q[6], tl0), 0.0f);
            float ey = fmaxf(fmaxf(q[10], th1) - fminf(q[7], tl1), 0.0f);
            float ez = fmaxf(fmaxf(q[11], th2) - fminf(q[8], tl2), 0.0f);
            float volc = ex * ey * ez;
            float giou = iou - (volc - uni) * fast_rcp(volc);

            float cost = COST_BBOX * l1 - COST_CLASS * acc[i] - COST_GIOU * giou;
            // nan_to_num: nan -> 0, +/-inf -> +/-FLT_MAX
            cost = (cost != cost) ? 0.0f : cost;
            cost = fminf(fmaxf(cost, -FLT_MAX), FLT_MAX);

            out[(size_t)(rowbase + r) * MT + m] = cost;
        }
    }
}

extern "C" void kernel_launch(void* const* d_in, const int* in_sizes, int n_in,
                              void* d_out, int out_size, void* d_ws, size_t ws_size,
                              hipStream_t stream) {
    const float* pred_logits = (const float*)d_in[0];
    const float* pred_boxes  = (const float*)d_in[1];
    const float* tgt_boxes   = (const float*)d_in[2];
    const int*   tgt_labels  = (const int*)d_in[3];
    float* out = (float*)d_out;

    const int nRows = in_sizes[1] / 6;   // bs*Q = 32768
    const int grid  = nRows / QT;        // 2048 blocks of 16 rows
    matcher_cost_kernel<<<grid, 256, 0, stream>>>(
        pred_logits, pred_boxes, tgt_boxes, tgt_labels, out);
}